// Match_52467320488264
// MI455X (gfx1250) — compile-verified
//
#include <hip/hip_runtime.h>
#include <stdint.h>

typedef float v2f __attribute__((ext_vector_type(2)));
typedef float v4f __attribute__((ext_vector_type(4)));
typedef float v8f __attribute__((ext_vector_type(8)));

#define L_DIM 4096
#define C_DIM 256
#define B_DIM 4
#define THRESH 0.2f
#define INV_TEMP 10.0f
#define KCH 32
#define NCH (C_DIM / KCH)   // 8 K-chunks

// ---- shared math helpers (must be bitwise-identical across kernels) ----
__device__ __forceinline__ float mval(float v, float rmx, float rsinv, float cmx, float csinv) {
    return (expf(v - rmx) * rsinv) * (expf(v - cmx) * csinv);
}
__device__ __forceinline__ void online_update(float& mx, float& sm, float v) {
    float nm = fmaxf(mx, v);
    sm = sm * expf(mx - nm) + expf(v - nm);
    mx = nm;
}
__device__ __forceinline__ void online_combine(float& mx, float& sm, float m2, float s2) {
    float nm = fmaxf(mx, m2);
    sm = sm * expf(mx - nm) + s2 * expf(m2 - nm);
    mx = nm;
}

// ---- 1) WMMA fp32 GEMM with async global->LDS double buffering ----
// M[n][l][s] = (f1[n,l,:] . f2[n,s,:]) / TEMP
__global__ __launch_bounds__(256) void gemm_kernel(const float* __restrict__ f1,
                                                   const float* __restrict__ f2,
                                                   float* __restrict__ M) {
    // pad 36 floats/row: row stride 144 B -> bank stride 4 (conflict-free b64 frag
    // reads) and every 16-B b128 async write lands 16-B aligned.
    __shared__ float As[2][64][36];
    __shared__ float Bs[2][64][36];
    __shared__ float Cs[64][64];
    const int t    = threadIdx.x;
    const int lane = t & 31;
    const int wave = t >> 5;
    const int r  = wave >> 1;      // row group 0..3 (16 rows each)
    const int cc = wave & 1;       // col group 0..1 (32 cols each)
    const int n  = blockIdx.z;
    const int m0 = blockIdx.x * 64;
    const int s0 = blockIdx.y * 64;

    const float* a_base = f1 + (size_t)(n * L_DIM + m0) * C_DIM;
    const float* b_base = f2 + (size_t)(n * L_DIM + s0) * C_DIM;

    // Each chunk = 64 rows x 32 K-floats per matrix = 512 b128 transfers
    // -> 2 async ops per thread per matrix (4 per wave per chunk total).
    auto issue_chunk = [&](int kc, int buf) {
#pragma unroll
        for (int it = 0; it < 2; ++it) {
            int idx = it * 256 + t;          // 0..511
            int rr  = idx >> 3;              // 0..63
            int qq  = (idx & 7) << 2;        // 0,4,...,28 (floats)
            unsigned lA = (unsigned)(uintptr_t)&As[buf][rr][qq];
            unsigned long long gA =
                (unsigned long long)(uintptr_t)(a_base + (size_t)rr * C_DIM + kc + qq);
            asm volatile("global_load_async_to_lds_b128 %0, %1, off"
                         :: "v"(lA), "v"(gA) : "memory");
            unsigned lB = (unsigned)(uintptr_t)&Bs[buf][rr][qq];
            unsigned long long gB =
                (unsigned long long)(uintptr_t)(b_base + (size_t)rr * C_DIM + kc + qq);
            asm volatile("global_load_async_to_lds_b128 %0, %1, off"
                         :: "v"(lB), "v"(gB) : "memory");
        }
    };

    v8f acc0 = {};
    v8f acc1 = {};
    const int fr  = (r << 4) + (lane & 15);          // A row (M index)
    const int fc0 = (cc << 5) + (lane & 15);         // B col (N index), frag0

    issue_chunk(0, 0);                                // prefetch chunk 0
    for (int c = 0; c < NCH; ++c) {
        const int buf = c & 1;
        if (c + 1 < NCH) {
            issue_chunk((c + 1) * KCH, (c + 1) & 1);  // prefetch next chunk
            // our 4 chunk-c ops finish before the 4 chunk-(c+1) ops (in-order)
            asm volatile("s_wait_asynccnt 4" ::: "memory");
        } else {
            asm volatile("s_wait_asynccnt 0" ::: "memory");
        }
        __syncthreads();                              // chunk c visible to all waves
#pragma unroll
        for (int kk = 0; kk < KCH; kk += 4) {
            int fk = kk + ((lane >> 4) << 1);         // K pair {0,1} or {2,3}
            v2f a  = *(const v2f*)&As[buf][fr][fk];
            v2f b0 = *(const v2f*)&Bs[buf][fc0][fk];
            v2f b1 = *(const v2f*)&Bs[buf][fc0 + 16][fk];
            acc0 = __builtin_amdgcn_wmma_f32_16x16x4_f32(false, a, false, b0,
                                                         (short)0, acc0, false, false);
            acc1 = __builtin_amdgcn_wmma_f32_16x16x4_f32(false, a, false, b1,
                                                         (short)0, acc1, false, false);
        }
        __syncthreads();                              // done reading buf before refill
    }

    // scatter accumulators into LDS C tile (scaled by 1/TEMP)
    {
        int mBase = (r << 4) + ((lane >> 4) << 3);    // rows v or v+8
        int nI = lane & 15;
#pragma unroll
        for (int v = 0; v < 8; ++v) {
            Cs[mBase + v][(cc << 5) + nI]      = acc0[v] * INV_TEMP;
            Cs[mBase + v][(cc << 5) + 16 + nI] = acc1[v] * INV_TEMP;
        }
    }
    __syncthreads();
    // coalesced 128b stores of the 64x64 tile
    float* mrow = M + (size_t)(n * L_DIM + m0) * L_DIM + s0;
#pragma unroll
    for (int it = 0; it < 4; ++it) {
        int lin = it * 256 + t;
        int rr  = lin >> 4;
        int q   = (lin & 15) << 2;
        *(v4f*)(mrow + (size_t)rr * L_DIM + q) = *(const v4f*)&Cs[rr][q];
    }
}

// ---- 2) per-row online softmax stats over s ----
__global__ __launch_bounds__(256) void row_stats_kernel(const float* __restrict__ M,
                                                        float* __restrict__ rmax,
                                                        float* __restrict__ rsum) {
    __shared__ float smx[256];
    __shared__ float ssm[256];
    const int t = threadIdx.x;
    const size_t row = blockIdx.x;
    const float* p = M + row * L_DIM;
    float mx = -INFINITY, sm = 0.f;
#pragma unroll
    for (int i = 0; i < 4; ++i) {
        v4f v = *(const v4f*)(p + (t << 2) + (i << 10));
        online_update(mx, sm, v.x);
        online_update(mx, sm, v.y);
        online_update(mx, sm, v.z);
        online_update(mx, sm, v.w);
    }
    smx[t] = mx; ssm[t] = sm;
    __syncthreads();
    for (int off = 128; off > 0; off >>= 1) {
        if (t < off) {
            float m0 = smx[t], s0 = ssm[t];
            online_combine(m0, s0, smx[t + off], ssm[t + off]);
            smx[t] = m0; ssm[t] = s0;
        }
        __syncthreads();
    }
    if (t == 0) { rmax[row] = smx[0]; rsum[row] = ssm[0]; }
}

// ---- 3) per-column online softmax stats, 256-row partials ----
__global__ __launch_bounds__(256) void col_stats_partial_kernel(const float* __restrict__ M,
                                                                float* __restrict__ pmax,
                                                                float* __restrict__ psum) {
    const int t = threadIdx.x;
    const int col = blockIdx.x * 256 + t;
    const int chunk = blockIdx.y;
    const int n = blockIdx.z;
    const float* p = M + (size_t)(n * L_DIM + chunk * 256) * L_DIM + col;
    float mx = -INFINITY, sm = 0.f;
    for (int r = 0; r < 256; ++r)
        online_update(mx, sm, p[(size_t)r * L_DIM]);
    size_t o = (size_t)(n * 16 + chunk) * L_DIM + col;
    pmax[o] = mx; psum[o] = sm;
}

// ---- 4) deterministic fixed-order combine of the 16 partials ----
__global__ __launch_bounds__(256) void col_stats_combine_kernel(const float* __restrict__ pmax,
                                                                const float* __restrict__ psum,
                                                                float* __restrict__ cmax,
                                                                float* __restrict__ csum) {
    const int i = blockIdx.x * 256 + threadIdx.x;     // 0..B*L-1
    const int n = i >> 12;
    const int s = i & (L_DIM - 1);
    float mx = -INFINITY, sm = 0.f;
#pragma unroll
    for (int ch = 0; ch < 16; ++ch) {
        size_t o = (size_t)(n * 16 + ch) * L_DIM + s;
        online_combine(mx, sm, pmax[o], psum[o]);
    }
    cmax[i] = mx; csum[i] = sm;
}

// ---- 5) column max of m, 256-row partials ----
__global__ __launch_bounds__(256) void mcol_partial_kernel(const float* __restrict__ M,
                                                           const float* __restrict__ rmax,
                                                           const float* __restrict__ rsum,
                                                           const float* __restrict__ cmax,
                                                           const float* __restrict__ csum,
                                                           float* __restrict__ mpart) {
    const int t = threadIdx.x;
    const int col = blockIdx.x * 256 + t;
    const int chunk = blockIdx.y;
    const int n = blockIdx.z;
    const float cmx   = cmax[n * L_DIM + col];
    const float csinv = 1.0f / csum[n * L_DIM + col];
    const float* p     = M + (size_t)(n * L_DIM + chunk * 256) * L_DIM + col;
    const float* rmx_p = rmax + n * L_DIM + chunk * 256;
    const float* rsm_p = rsum + n * L_DIM + chunk * 256;
    float mm = -INFINITY;
    for (int r = 0; r < 256; ++r) {
        float m = mval(p[(size_t)r * L_DIM], rmx_p[r], 1.0f / rsm_p[r], cmx, csinv);
        mm = fmaxf(mm, m);
    }
    mpart[(size_t)(n * 16 + chunk) * L_DIM + col] = mm;
}

__global__ __launch_bounds__(256) void mcol_combine_kernel(const float* __restrict__ mpart,
                                                           float* __restrict__ mcolmax) {
    const int i = blockIdx.x * 256 + threadIdx.x;
    const int n = i >> 12;
    const int s = i & (L_DIM - 1);
    float mm = -INFINITY;
#pragma unroll
    for (int ch = 0; ch < 16; ++ch)
        mm = fmaxf(mm, mpart[(size_t)(n * 16 + ch) * L_DIM + s]);
    mcolmax[i] = mm;
}

// ---- 7) row scan: mutual mask + first-match index ----
__global__ __launch_bounds__(256) void row_scan_kernel(const float* __restrict__ M,
                                                       const float* __restrict__ rmax,
                                                       const float* __restrict__ rsum,
                                                       const float* __restrict__ cmax,
                                                       const float* __restrict__ csum,
                                                       const float* __restrict__ mcolmax,
                                                       int* __restrict__ jidx) {
    __shared__ float mrow[L_DIM];
    __shared__ float red[256];
    __shared__ int redi[256];
    const int t = threadIdx.x;
    const size_t row = blockIdx.x;
    const int n = (int)(row >> 12);
    const float rmx   = rmax[row];
    const float rsinv = 1.0f / rsum[row];
    const float* p     = M + row * L_DIM;
    const float* cmx_p = cmax + n * L_DIM;
    const float* csm_p = csum + n * L_DIM;
    float mm = -INFINITY;
#pragma unroll
    for (int i = 0; i < 16; ++i) {
        int s = t + (i << 8);
        float m = mval(p[s], rmx, rsinv, cmx_p[s], 1.0f / csm_p[s]);
        mrow[s] = m;
        mm = fmaxf(mm, m);
    }
    red[t] = mm;
    __syncthreads();
    for (int off = 128; off > 0; off >>= 1) {
        if (t < off) red[t] = fmaxf(red[t], red[t + off]);
        __syncthreads();
    }
    const float mrmax = red[0];
    const float* mcm_p = mcolmax + n * L_DIM;
    int best = 0x7fffffff;
#pragma unroll
    for (int i = 0; i < 16; ++i) {
        int s = t + (i << 8);
        float m = mrow[s];
        if (m > THRESH && m == mrmax && m == mcm_p[s] && s < best) best = s;
    }
    redi[t] = best;
    __syncthreads();
    for (int off = 128; off > 0; off >>= 1) {
        if (t < off) redi[t] = min(redi[t], redi[t + off]);
        __syncthreads();
    }
    if (t == 0) jidx[row] = (redi[0] == 0x7fffffff) ? -1 : redi[0];
}

// ---- 8) gather-subtract + NCHW transpose through LDS ----
__global__ __launch_bounds__(256) void output_kernel(const float* __restrict__ f1,
                                                     const float* __restrict__ f2,
                                                     const int* __restrict__ jidx,
                                                     float* __restrict__ out) {
    __shared__ float Ts[64][65];
    __shared__ int js[64];
    const int t  = threadIdx.x;
    const int n  = blockIdx.z;
    const int l0 = blockIdx.x * 64;
    const int c0 = blockIdx.y * 64;
    if (t < 64) js[t] = jidx[n * L_DIM + l0 + t];
    __syncthreads();
#pragma unroll
    for (int it = 0; it < 16; ++it) {
        int lin = it * 256 + t;
        int lr  = lin >> 6;        // l within tile
        int ccc = lin & 63;        // channel within tile
        int jv = js[lr];
        float val = f1[(size_t)(n * L_DIM + l0 + lr) * C_DIM + c0 + ccc];
        if (jv >= 0) val -= f2[(size_t)(n * L_DIM + jv) * C_DIM + c0 + ccc];
        Ts[ccc][lr] = val;
    }
    __syncthreads();
#pragma unroll
    for (int it = 0; it < 16; ++it) {
        int lin = it * 256 + t;
        int chn = lin >> 6;
        int ll  = lin & 63;
        out[(size_t)(n * C_DIM + c0 + chn) * L_DIM + l0 + ll] = Ts[chn][ll];
    }
}

extern "C" void kernel_launch(void* const* d_in, const int* in_sizes, int n_in,
                              void* d_out, int out_size, void* d_ws, size_t ws_size,
                              hipStream_t stream) {
    const float* f1 = (const float*)d_in[0];
    const float* f2 = (const float*)d_in[1];
    float* out = (float*)d_out;
    char* ws = (char*)d_ws;

    const size_t BL = (size_t)B_DIM * L_DIM;                 // 16384
    float* M = (float*)ws;                                   // 268,435,456 B
    size_t ofs = BL * L_DIM * sizeof(float);
    float* rmax    = (float*)(ws + ofs); ofs += BL * 4;
    float* rsum    = (float*)(ws + ofs); ofs += BL * 4;
    float* cmax    = (float*)(ws + ofs); ofs += BL * 4;
    float* csum    = (float*)(ws + ofs); ofs += BL * 4;
    float* mcolmax = (float*)(ws + ofs); ofs += BL * 4;
    int*   jidx    = (int*)(ws + ofs);   ofs += BL * 4;
    float* pmax    = (float*)(ws + ofs); ofs += BL * 16 * 4;
    float* psum    = (float*)(ws + ofs); ofs += BL * 16 * 4;
    float* mpart   = (float*)(ws + ofs); ofs += BL * 16 * 4;

    dim3 gGemm(L_DIM / 64, L_DIM / 64, B_DIM);
    gemm_kernel<<<gGemm, 256, 0, stream>>>(f1, f2, M);

    row_stats_kernel<<<B_DIM * L_DIM, 256, 0, stream>>>(M, rmax, rsum);

    dim3 gCol(L_DIM / 256, L_DIM / 256, B_DIM);
    col_stats_partial_kernel<<<gCol, 256, 0, stream>>>(M, pmax, psum);
    col_stats_combine_kernel<<<(B_DIM * L_DIM) / 256, 256, 0, stream>>>(pmax, psum, cmax, csum);

    mcol_partial_kernel<<<gCol, 256, 0, stream>>>(M, rmax, rsum, cmax, csum, mpart);
    mcol_combine_kernel<<<(B_DIM * L_DIM) / 256, 256, 0, stream>>>(mpart, mcolmax);

    row_scan_kernel<<<B_DIM * L_DIM, 256, 0, stream>>>(M, rmax, rsum, cmax, csum, mcolmax, jidx);

    dim3 gOut(L_DIM / 64, C_DIM / 64, B_DIM);
    output_kernel<<<gOut, 256, 0, stream>>>(f1, f2, jidx, out);
}